// Aggregator_84696755077585
// MI455X (gfx1250) — compile-verified
//
#include <hip/hip_runtime.h>

typedef __attribute__((ext_vector_type(2))) float v2f;
typedef __attribute__((ext_vector_type(8))) float v8f;

#define N_ENT   100000
#define N_USR   4096
#define N_ITEM  8192
#define N_REL   32
#define N_CLS   3
#define D_DIM   128
#define N_EDGES 800000
#define NNZ     204800

// D = A(16xK rowmajor) * B(16xK rowmajor)^T accumulated 4 K-steps at a time.
// f32 WMMA per-lane layout (ISA 7.12.2): A vgpr r holds A[lane%16][2*(lane/16)+r],
// B mirrors with its own row index; C/D vgpr r holds (M = r + 8*(lane/16), N = lane%16).
__device__ __forceinline__ v8f wmma4(v2f a, v2f b, v8f c) {
  return __builtin_amdgcn_wmma_f32_16x16x4_f32(
      /*neg_a=*/false, a, /*neg_b=*/false, b,
      /*c_mod=*/(short)0, c, /*reuse_a=*/false, /*reuse_b=*/false);
}

// ---- tiny prep kernels -----------------------------------------------------

__global__ void k_rel_sum(const float* __restrict__ rel, float* __restrict__ rel_sum) {
  int d = threadIdx.x;
  float s = 0.f;
#pragma unroll
  for (int r = 0; r < N_REL; ++r) s += rel[r * D_DIM + d];
  rel_sum[d] = s;
}

// b2t[d][i] = item_emb[i][d] * rel_sum[d]   (transposed, K-major rows for WMMA B)
__global__ void k_b2t(const float* __restrict__ item, const float* __restrict__ rel_sum,
                      float* __restrict__ b2t) {
  int t = blockIdx.x * blockDim.x + threadIdx.x;
  if (t >= D_DIM * N_ITEM) return;
  int d = t / N_ITEM;
  int i = t - d * N_ITEM;
  b2t[t] = item[(size_t)i * D_DIM + d] * rel_sum[d];
}

// user_cls_att = softmax(user_emb @ usr_cls_w^T) over 3 classes
__global__ void k_user_att(const float* __restrict__ user, const float* __restrict__ w,
                           float* __restrict__ att) {
  int u = blockIdx.x * blockDim.x + threadIdx.x;
  if (u >= N_USR) return;
  const float* ur = user + (size_t)u * D_DIM;
  float l0 = 0.f, l1 = 0.f, l2 = 0.f;
#pragma unroll 4
  for (int k = 0; k < D_DIM; ++k) {
    float x = ur[k];
    l0 += x * w[k];
    l1 += x * w[D_DIM + k];
    l2 += x * w[2 * D_DIM + k];
  }
  float m = fmaxf(l0, fmaxf(l1, l2));
  float e0 = __expf(l0 - m), e1 = __expf(l1 - m), e2 = __expf(l2 - m);
  float inv = 1.f / (e0 + e1 + e2);
  att[u * 3 + 0] = e0 * inv;
  att[u * 3 + 1] = e1 * inv;
  att[u * 3 + 2] = e2 * inv;
}

// ---- WMMA: logits = entity_emb @ relation_emb^T  ([100000,128]x[128,32]) ----

__global__ void k_logits(const float* __restrict__ ent, const float* __restrict__ rel,
                         float* __restrict__ logits) {
  int lane = threadIdx.x & 31;
  int tile = blockIdx.x * (blockDim.x >> 5) + (threadIdx.x >> 5);
  if (tile >= N_ENT / 16) return;           // whole wave exits together (EXEC stays all-1)
  int m0   = tile << 4;
  int row  = lane & 15;
  int koff = (lane >> 4) << 1;              // 0 or 2
  const float* ap = ent + (size_t)(m0 + row) * D_DIM + koff;
  const float* b0 = rel + (size_t)row * D_DIM + koff;          // relations 0..15
  const float* b1 = rel + (size_t)(row + 16) * D_DIM + koff;   // relations 16..31
  v8f c0 = {}; v8f c1 = {};
#pragma unroll 4
  for (int k = 0; k < D_DIM; k += 4) {
    v2f a  = *(const v2f*)(ap + k);
    v2f x0 = *(const v2f*)(b0 + k);
    v2f x1 = *(const v2f*)(b1 + k);
    c0 = wmma4(a, x0, c0);
    c1 = wmma4(a, x1, c1);
  }
  int mbase = m0 + ((lane >> 4) << 3);
  int col   = lane & 15;
#pragma unroll
  for (int r = 0; r < 8; ++r) {
    logits[(size_t)(mbase + r) * N_REL + col]      = c0[r];
    logits[(size_t)(mbase + r) * N_REL + 16 + col] = c1[r];
  }
}

__global__ void k_softmax32(float* __restrict__ logits) {
  int e = blockIdx.x * blockDim.x + threadIdx.x;
  if (e >= N_ENT) return;
  float* p = logits + (size_t)e * N_REL;
  float v[N_REL];
  float m = -1e30f;
#pragma unroll
  for (int r = 0; r < N_REL; ++r) { v[r] = p[r]; m = fmaxf(m, v[r]); }
  float s = 0.f;
#pragma unroll
  for (int r = 0; r < N_REL; ++r) { v[r] = __expf(v[r] - m); s += v[r]; }
  float inv = 1.f / s;
#pragma unroll
  for (int r = 0; r < N_REL; ++r) p[r] = v[r] * inv;
}

// ---- edge message scatter: one wave per edge, float4 per lane --------------

__global__ void k_edge_scatter(const float* __restrict__ ent, const float* __restrict__ rel,
                               const int* __restrict__ edge_index, const int* __restrict__ etype,
                               const float* __restrict__ imp, const float* __restrict__ smax,
                               float* __restrict__ entity_agg) {
  long long g = (long long)blockIdx.x * blockDim.x + threadIdx.x;
  int e    = (int)(g >> 5);
  int lane = (int)(g & 31);
  if (e >= N_EDGES) return;
  int h = edge_index[e];
  int t = edge_index[N_EDGES + e];
  int r = etype[e];
  float s = smax[(size_t)h * N_REL + r] * imp[e];
  float4 tv = ((const float4*)(ent + (size_t)t * D_DIM))[lane];
  float4 rv = ((const float4*)(rel + (size_t)r * D_DIM))[lane];
  float* out = entity_agg + (size_t)h * D_DIM + lane * 4;
  atomicAdd(out + 0, tv.x * rv.x * s);
  atomicAdd(out + 1, tv.y * rv.y * s);
  atomicAdd(out + 2, tv.z * rv.z * s);
  atomicAdd(out + 3, tv.w * rv.w * s);
}

// ---- COO interact_mat @ entity_emb scatter ---------------------------------

__global__ void k_user_scatter(const float* __restrict__ ent, const int* __restrict__ rows,
                               const int* __restrict__ cols, const float* __restrict__ vals,
                               float* __restrict__ user_agg) {
  long long g = (long long)blockIdx.x * blockDim.x + threadIdx.x;
  int e    = (int)(g >> 5);
  int lane = (int)(g & 31);
  if (e >= NNZ) return;
  int u = rows[e];
  int c = cols[e];
  float v = vals[e];
  float4 cv = ((const float4*)(ent + (size_t)c * D_DIM))[lane];
  float* out = user_agg + (size_t)u * D_DIM + lane * 4;
  atomicAdd(out + 0, cv.x * v);
  atomicAdd(out + 1, cv.y * v);
  atomicAdd(out + 2, cv.z * v);
  atomicAdd(out + 3, cv.w * v);
}

// ---- the big one: sum_c att[u,c] * (inter_cls_mat[c] @ item_emb_2)[u,:] ----
// Block = 8 waves = the 8 N-tiles of one 16-user M-tile (A rows shared via L0/L2).
// Each wave keeps 3 class accumulators sharing one B fetch per K-step.

__global__ void k_disen(const float* __restrict__ icm, const float* __restrict__ b2t,
                        const float* __restrict__ att, float* __restrict__ user_agg) {
  int lane = threadIdx.x & 31;
  int n0   = (threadIdx.x >> 5) << 4;   // wave -> N-tile (D columns)
  int m0   = blockIdx.x << 4;           // block -> M-tile (users)
  int row  = lane & 15;
  int koff = (lane >> 4) << 1;
  const size_t K = N_ITEM;
  const float* a0 = icm + (size_t)(m0 + row) * K + koff;            // class 0
  const float* a1 = a0 + (size_t)N_USR * K;                          // class 1
  const float* a2 = a1 + (size_t)N_USR * K;                          // class 2
  const float* bp = b2t + (size_t)(n0 + row) * K + koff;             // B^T rows
  v8f c0 = {}; v8f c1 = {}; v8f c2 = {};
#pragma unroll 4
  for (int k = 0; k < N_ITEM; k += 4) {
    v2f b  = *(const v2f*)(bp + k);
    v2f x0 = *(const v2f*)(a0 + k);
    v2f x1 = *(const v2f*)(a1 + k);
    v2f x2 = *(const v2f*)(a2 + k);
    c0 = wmma4(x0, b, c0);
    c1 = wmma4(x1, b, c1);
    c2 = wmma4(x2, b, c2);
  }
  int ubase = m0 + ((lane >> 4) << 3);
  int d     = n0 + (lane & 15);
#pragma unroll
  for (int r = 0; r < 8; ++r) {
    int u = ubase + r;
    float w0 = att[u * 3 + 0], w1 = att[u * 3 + 1], w2 = att[u * 3 + 2];
    // exclusive ownership of (u,d): plain RMW, runs after k_user_scatter in stream order
    user_agg[(size_t)u * D_DIM + d] += w0 * c0[r] + w1 * c1[r] + w2 * c2[r];
  }
}

// ---- launch ----------------------------------------------------------------

extern "C" void kernel_launch(void* const* d_in, const int* in_sizes, int n_in,
                              void* d_out, int out_size, void* d_ws, size_t ws_size,
                              hipStream_t stream) {
  const float* entity_emb    = (const float*)d_in[0];
  const float* item_emb      = (const float*)d_in[1];
  const float* user_emb      = (const float*)d_in[2];
  // d_in[3] latent_emb: unused by reference
  const float* relation_emb  = (const float*)d_in[4];
  const int*   edge_index    = (const int*)d_in[5];
  const int*   edge_type     = (const int*)d_in[6];
  const float* edge_imp      = (const float*)d_in[7];
  const int*   inter_rows    = (const int*)d_in[8];
  const int*   inter_cols    = (const int*)d_in[9];
  const float* inter_vals    = (const float*)d_in[10];
  // d_in[11] disen_weight_att, d_in[12] ent_rel_w: unused by reference
  const float* usr_cls_w     = (const float*)d_in[13];
  const float* inter_cls_mat = (const float*)d_in[14];

  float* entity_agg = (float*)d_out;
  float* user_agg   = entity_agg + (size_t)N_ENT * D_DIM;

  float* ws      = (float*)d_ws;
  float* logits  = ws;                                   // N_ENT*N_REL   (3.2M f)
  float* b2t     = logits + (size_t)N_ENT * N_REL;       // D_DIM*N_ITEM  (1.05M f)
  float* att     = b2t + (size_t)D_DIM * N_ITEM;         // N_USR*N_CLS
  float* rel_sum = att + (size_t)N_USR * N_CLS;          // D_DIM

  hipMemsetAsync(d_out, 0, (size_t)out_size * sizeof(float), stream);

  k_rel_sum<<<1, 128, 0, stream>>>(relation_emb, rel_sum);
  k_b2t<<<(D_DIM * N_ITEM + 255) / 256, 256, 0, stream>>>(item_emb, rel_sum, b2t);
  k_user_att<<<(N_USR + 255) / 256, 256, 0, stream>>>(user_emb, usr_cls_w, att);
  k_logits<<<(N_ENT / 16 + 7) / 8, 256, 0, stream>>>(entity_emb, relation_emb, logits);
  k_softmax32<<<(N_ENT + 255) / 256, 256, 0, stream>>>(logits);
  k_edge_scatter<<<(N_EDGES * 32) / 256, 256, 0, stream>>>(entity_emb, relation_emb,
      edge_index, edge_type, edge_imp, logits, entity_agg);
  k_user_scatter<<<(NNZ * 32) / 256, 256, 0, stream>>>(entity_emb, inter_rows, inter_cols,
      inter_vals, user_agg);
  k_disen<<<N_USR / 16, 256, 0, stream>>>(inter_cls_mat, b2t, att, user_agg);
}